// Recommender_79216376807729
// MI455X (gfx1250) — compile-verified
//
#include <hip/hip_runtime.h>

#define N_ENT  200000
#define N_USR  100000
#define D      64
#define N_EDGE 1000000
#define NNZ    1000000
#define EPSF   1e-12f

typedef __attribute__((ext_vector_type(2))) float v2f;
typedef __attribute__((ext_vector_type(8))) float v8f;

// ---- order-preserving float<->uint encoding for exact atomic segment-max ----
__device__ __forceinline__ unsigned enc_f32(float f) {
    unsigned u = __float_as_uint(f);
    return (u & 0x80000000u) ? ~u : (u | 0x80000000u);
}
__device__ __forceinline__ float dec_f32(unsigned u) {
    unsigned b = (u & 0x80000000u) ? (u & 0x7FFFFFFFu) : ~u;
    return __uint_as_float(b);
}
#define ENC_NEG_INF 0x007FFFFFu   // enc(-inf)

__global__ void fill_u32_k(unsigned* __restrict__ p, unsigned v, long long n) {
    long long i = (long long)blockIdx.x * blockDim.x + threadIdx.x;
    if (i < n) p[i] = v;
}

// ---- KG edge scores via V_WMMA_F32_16X16X4_F32: one wave handles 16 edges ----
// score_i = sum_d eh_i[d]*r_i[d]*et_i[d] = diag(A x B^T), A=eh.*r, B=et (16x64 fp32)
__global__ void kg_score_wmma_k(const float* __restrict__ ent,
                                const float* __restrict__ rel,
                                const int*   __restrict__ head,
                                const int*   __restrict__ tail,
                                const int*   __restrict__ etype,
                                float*       __restrict__ score) {
    const int lane = threadIdx.x & 31;
    const long long wave = (long long)blockIdx.x * (blockDim.x >> 5) + (threadIdx.x >> 5);
    const long long base = wave * 16;
    if (base >= N_EDGE) return;                    // uniform per wave

    const int sub = lane & 15;                     // A: M index / B: N index (edge in group)
    const int hi  = lane >> 4;                     // selects K pair within K=4 block
    long long e = base + sub;
    if (e >= N_EDGE) e = N_EDGE - 1;               // clamp loads; stores guarded below

    const int h  = head[e];
    const int t  = tail[e];
    const int rt = etype[e] - 1;
    const float* eh = ent + (long long)h  * D;
    const float* et = ent + (long long)t  * D;
    const float* rr = rel + (long long)rt * D;

    v8f c = {0.f, 0.f, 0.f, 0.f, 0.f, 0.f, 0.f, 0.f};
#pragma unroll
    for (int kb = 0; kb < 16; ++kb) {              // 16 x (K=4) = K=64
        const int ko = kb * 4 + hi * 2;
        v2f a, b;
        a.x = eh[ko]     * rr[ko];
        a.y = eh[ko + 1] * rr[ko + 1];
        b.x = et[ko];
        b.y = et[ko + 1];
        c = __builtin_amdgcn_wmma_f32_16x16x4_f32(false, a, false, b,
                                                  (short)0, c, false, false);
    }
    // Diagonal extraction per C/D layout: (i,i) i<8 -> lane i, c[i];
    // i in 8..15 -> lane i+16 (24..31), c[i-8].
    float s = 0.f;
#pragma unroll
    for (int v = 0; v < 8; ++v) {
        if (lane == v)      s = c[v];
        if (lane == 24 + v) s = c[v];
    }
    long long oi = -1;
    if (lane < 8)        oi = base + lane;
    else if (lane >= 24) oi = base + 8 + (lane - 24);
    if (oi >= 0 && oi < N_EDGE) score[oi] = s;
}

__global__ void seg_max_k(const float* __restrict__ score,
                          const int* __restrict__ head,
                          unsigned* __restrict__ segenc) {
    long long i = (long long)blockIdx.x * blockDim.x + threadIdx.x;
    if (i >= N_EDGE) return;
    atomicMax(&segenc[head[i]], enc_f32(score[i]));
}

__global__ void edge_exp_k(float* __restrict__ score,
                           const int* __restrict__ head,
                           const unsigned* __restrict__ segenc,
                           float* __restrict__ denom) {
    long long i = (long long)blockIdx.x * blockDim.x + threadIdx.x;
    if (i >= N_EDGE) return;
    const int h = head[i];
    float m = dec_f32(segenc[h]);
    if (!isfinite(m)) m = 0.f;                     // reference: non-finite seg_max -> 0
    float ex = expf(score[i] - m);
    score[i] = ex;                                 // keep exp; divide folded into hop
    atomicAdd(&denom[h], ex);
}

// ---- one KG hop: e_next[head] += (exp/denom) * rel * e_cur[tail]  (wave/edge) ----
__global__ void kg_hop_scatter_k(const float* __restrict__ ecur,
                                 const float* __restrict__ rel,
                                 const int* __restrict__ head,
                                 const int* __restrict__ tail,
                                 const int* __restrict__ etype,
                                 const float* __restrict__ escore,
                                 const float* __restrict__ denom,
                                 float* __restrict__ enext) {
    const int lane = threadIdx.x & 31;
    const long long e = (long long)blockIdx.x * (blockDim.x >> 5) + (threadIdx.x >> 5);
    if (e >= N_EDGE) return;
    const int h  = head[e];
    const int t  = tail[e];
    const int rt = etype[e] - 1;
    const float w = escore[e] / (denom[h] + EPSF);
    const float* er = ecur + (long long)t  * D;
    const float* rr = rel  + (long long)rt * D;
    float* out = enext + (long long)h * D;
    const int d0 = lane, d1 = lane + 32;
    atomicAdd(&out[d0], w * rr[d0] * er[d0]);
    atomicAdd(&out[d1], w * rr[d1] * er[d1]);
}

// ---- L2-normalize each row of agg into cur, and outacc += normalized ----
__global__ void normalize_accum_k(const float* __restrict__ agg,
                                  float* __restrict__ cur,
                                  float* __restrict__ outacc,
                                  int nrows) {
    const int lane = threadIdx.x & 31;
    const long long r = (long long)blockIdx.x * (blockDim.x >> 5) + (threadIdx.x >> 5);
    if (r >= nrows) return;
    const float* src = agg + r * (long long)D;
    float v0 = src[lane], v1 = src[lane + 32];
    float ss = v0 * v0 + v1 * v1;
#pragma unroll
    for (int off = 16; off > 0; off >>= 1) ss += __shfl_xor(ss, off, 32);
    const float inv = 1.0f / fmaxf(sqrtf(ss), EPSF);
    const float n0 = v0 * inv, n1 = v1 * inv;
    float* dc = cur    + r * (long long)D;
    float* da = outacc + r * (long long)D;
    dc[lane] = n0;        dc[lane + 32] = n1;
    da[lane] += n0;       da[lane + 32] += n1;
}

// ---- one LightGCN hop: both directions in a single pass (wave per nnz) ----
__global__ void cf_scatter_k(const float* __restrict__ ucur,
                             const float* __restrict__ ecur,
                             const int* __restrict__ row,
                             const int* __restrict__ col,
                             const float* __restrict__ val,
                             float* __restrict__ eagg,
                             float* __restrict__ uagg) {
    const int lane = threadIdx.x & 31;
    const long long i = (long long)blockIdx.x * (blockDim.x >> 5) + (threadIdx.x >> 5);
    if (i >= NNZ) return;
    const int r = row[i];
    const int c = col[i];
    const float v = val[i];
    const float* u = ucur + (long long)r * D;
    const float* e = ecur + (long long)c * D;
    float* ea = eagg + (long long)c * D;
    float* ua = uagg + (long long)r * D;
    const int d0 = lane, d1 = lane + 32;
    atomicAdd(&ea[d0], v * u[d0]);
    atomicAdd(&ea[d1], v * u[d1]);
    atomicAdd(&ua[d0], v * e[d0]);
    atomicAdd(&ua[d1], v * e[d1]);
}

extern "C" void kernel_launch(void* const* d_in, const int* in_sizes, int n_in,
                              void* d_out, int out_size, void* d_ws, size_t ws_size,
                              hipStream_t stream) {
    (void)in_sizes; (void)n_in; (void)out_size; (void)ws_size;
    const float* ent   = (const float*)d_in[0];
    const float* usr   = (const float*)d_in[1];
    const float* rel   = (const float*)d_in[2];
    const int*   eidx  = (const int*)d_in[3];     // [2, E]
    const int*   etype = (const int*)d_in[4];
    const int*   iidx  = (const int*)d_in[5];     // [2, NNZ]
    const float* ival  = (const float*)d_in[6];
    const int* head = eidx;
    const int* tail = eidx + N_EDGE;
    const int* irow = iidx;
    const int* icol = iidx + NNZ;

    float* out_kg = (float*)d_out;
    float* out_cf = out_kg + (size_t)N_ENT * D;
    float* out_u  = out_cf + (size_t)N_ENT * D;

    char* wsp = (char*)d_ws;
    auto wsalloc = [&](size_t bytes) -> char* {
        char* p = wsp;
        wsp += (bytes + 255) & ~(size_t)255;
        return p;
    };
    float*    escore = (float*)   wsalloc(sizeof(float)    * (size_t)N_EDGE);
    unsigned* segenc = (unsigned*)wsalloc(sizeof(unsigned) * (size_t)N_ENT);
    float*    denom  = (float*)   wsalloc(sizeof(float)    * (size_t)N_ENT);
    float*    bufA   = (float*)   wsalloc(sizeof(float)    * (size_t)N_ENT * D);
    float*    bufB   = (float*)   wsalloc(sizeof(float)    * (size_t)N_ENT * D);
    float*    bufUA  = (float*)   wsalloc(sizeof(float)    * (size_t)N_USR * D);
    float*    bufUB  = (float*)   wsalloc(sizeof(float)    * (size_t)N_USR * D);

    const int TB = 256;                           // 8 wave32 per block

    // Output accumulators start at the raw embeddings.
    hipMemcpyAsync(out_kg, ent, sizeof(float) * (size_t)N_ENT * D, hipMemcpyDeviceToDevice, stream);
    hipMemcpyAsync(out_cf, ent, sizeof(float) * (size_t)N_ENT * D, hipMemcpyDeviceToDevice, stream);
    hipMemcpyAsync(out_u,  usr, sizeof(float) * (size_t)N_USR * D, hipMemcpyDeviceToDevice, stream);

    // 1) KG attention scores (WMMA): wave per 16 edges
    {
        long long waves  = (N_EDGE + 15) / 16;
        long long blocks = (waves + 7) / 8;
        kg_score_wmma_k<<<(unsigned)blocks, TB, 0, stream>>>(ent, rel, head, tail, etype, escore);
    }
    // 2) exact scatter-softmax pieces
    fill_u32_k<<<(N_ENT + TB - 1) / TB, TB, 0, stream>>>(segenc, ENC_NEG_INF, N_ENT);
    seg_max_k<<<(N_EDGE + TB - 1) / TB, TB, 0, stream>>>(escore, head, segenc);
    hipMemsetAsync(denom, 0, sizeof(float) * (size_t)N_ENT, stream);
    edge_exp_k<<<(N_EDGE + TB - 1) / TB, TB, 0, stream>>>(escore, head, segenc, denom);

    // 3) KG hops
    {
        const float* ecur = ent;
        for (int hop = 0; hop < 2; ++hop) {
            hipMemsetAsync(bufA, 0, sizeof(float) * (size_t)N_ENT * D, stream);
            kg_hop_scatter_k<<<(unsigned)((N_EDGE + 7) / 8), TB, 0, stream>>>(
                ecur, rel, head, tail, etype, escore, denom, bufA);
            normalize_accum_k<<<(unsigned)(((long long)N_ENT + 7) / 8), TB, 0, stream>>>(
                bufA, bufB, out_kg, N_ENT);
            ecur = bufB;
        }
    }

    // 4) LightGCN hops
    {
        const float* ucur = usr;
        const float* ecur = ent;
        for (int hop = 0; hop < 2; ++hop) {
            hipMemsetAsync(bufA,  0, sizeof(float) * (size_t)N_ENT * D, stream);
            hipMemsetAsync(bufUA, 0, sizeof(float) * (size_t)N_USR * D, stream);
            cf_scatter_k<<<(unsigned)((NNZ + 7) / 8), TB, 0, stream>>>(
                ucur, ecur, irow, icol, ival, bufA, bufUA);
            normalize_accum_k<<<(unsigned)(((long long)N_ENT + 7) / 8), TB, 0, stream>>>(
                bufA, bufB, out_cf, N_ENT);
            normalize_accum_k<<<(unsigned)(((long long)N_USR + 7) / 8), TB, 0, stream>>>(
                bufUA, bufUB, out_u, N_USR);
            ecur = bufB;
            ucur = bufUB;
        }
    }
}